// RCL_49228915147432
// MI455X (gfx1250) — compile-verified
//
#include <hip/hip_runtime.h>
#include <math.h>

// ---------------- problem constants (match reference) ----------------
#define B_    512
#define NS_   2
#define S_    128
#define H_    768
#define DS_   384
#define NL_   40
#define H3_   (3 * H_)
#define TEMP_ 0.05f
#define ALPHA_ 0.15f
#define GAMMA_ 7.5f
#define EPS_  1e-8f

typedef __attribute__((ext_vector_type(2))) float v2f;
typedef __attribute__((ext_vector_type(4))) float v4f;
typedef __attribute__((ext_vector_type(8))) float v8f;

// ---------------- wave / block reductions (wave32) ----------------
__device__ __forceinline__ float waveMax(float v) {
#pragma unroll
    for (int o = 16; o; o >>= 1) v = fmaxf(v, __shfl_xor(v, o, 32));
    return v;
}
__device__ __forceinline__ float waveSum(float v) {
#pragma unroll
    for (int o = 16; o; o >>= 1) v += __shfl_xor(v, o, 32);
    return v;
}
__device__ __forceinline__ float blockMax(float v, float* sm) {
    int t = threadIdx.x, nw = blockDim.x >> 5;
    v = waveMax(v);
    __syncthreads();
    if ((t & 31) == 0) sm[t >> 5] = v;
    __syncthreads();
    float r = sm[0];
    for (int w = 1; w < nw; ++w) r = fmaxf(r, sm[w]);
    return r;
}
__device__ __forceinline__ float blockSum(float v, float* sm) {
    int t = threadIdx.x, nw = blockDim.x >> 5;
    v = waveSum(v);
    __syncthreads();
    if ((t & 31) == 0) sm[t >> 5] = v;
    __syncthreads();
    float r = sm[0];
    for (int w = 1; w < nw; ++w) r += sm[w];
    return r;
}

// ---------------- init ----------------
__global__ void init_out_kernel(float* out) {
    if (threadIdx.x == 0 && blockIdx.x == 0) out[0] = 0.0f;
}

// ---------------- gather + tanh prep ----------------
__global__ void gather_prep_kernel(const float* __restrict__ bert,
                                   const int* __restrict__ eidx,
                                   float* __restrict__ RH,
                                   float* __restrict__ TOKIN) {
    const int b = blockIdx.x;
    const float* s0 = bert + (size_t)b * NS_ * S_ * H_;
    const float* s1 = s0 + (size_t)S_ * H_;
    const int i00 = eidx[b * 4 + 0];
    const int i01 = eidx[b * 4 + 1];
    const int i10 = eidx[b * 4 + 2];
    const int i11 = eidx[b * 4 + 3];
    float* rh = RH + (size_t)b * H3_;
    float* tk = TOKIN + (size_t)b * 4 * H_;
    for (int h = threadIdx.x; h < H_; h += blockDim.x) {
        float cls  = s0[h];
        float head = s0[(size_t)i00 * H_ + h];
        float tail = s0[(size_t)i01 * H_ + h];
        float h2a  = s1[(size_t)i10 * H_ + h];
        float h2b  = s1[(size_t)i11 * H_ + h];
        rh[h]          = tanhf(cls);
        rh[H_ + h]     = tanhf(head);
        rh[2 * H_ + h] = tanhf(tail);
        tk[h]          = head;
        tk[H_ + h]     = tail;
        tk[2 * H_ + h] = h2a;
        tk[3 * H_ + h] = h2b;
    }
}

// ---------------- LDS-staged f32 WMMA GEMM ----------------
// C[M,N] = f( A[M,K](lda) x B[K,N](ldb) + bias )   (BT: B stored N x K row-major)
// MODE 0: +bias, 1: tanh(+bias), 2: raw (bias unused)
// Block: 128 threads = 4 waves. Block tile: 128(M) x 64(N).
// Each wave: two 16-row M tiles x 64 N (8 f32 accumulators = 64 VGPRs).
//
// B chunk (KC_=32 K x 64 N) is staged in LDS in *fragment order*: for K-group
// p (4 K values) and column n, the pair {B[4p+s][n], B[4p+2+s][n]} (s = lane
// half) is contiguous & 8B aligned, so each WMMA B operand is ONE ds_load_b64
// straight into an even-aligned VGPR pair -- no register shuffles, and all
// inner-loop LDS addresses are lane-base + constant immediate offsets.
// Half-wave rows are skewed by 160 floats so the two halves hit disjoint banks.
// M must be a multiple of 128, K a multiple of 32.
//
// Fragment layout assumption (V_WMMA_F32_16X16X4_F32, wave32):
//   A 16x4: lanes 0-15 rows M, VGPR{0,1} = K{0,1}; lanes 16-31 = K{2,3}
//   B 4x16: VGPR0 = K rows {0,1} (half-wave split), VGPR1 = K rows {2,3}
#define KC_ 32
#define PKSTRIDE_ 320   // floats per K-group of 4: two skewed 160-float s-rows

template <int MODE, bool BT>
__global__ void __launch_bounds__(128)
gemm_wmma_f32(const float* __restrict__ A, int lda,
              const float* __restrict__ Bm, int ldb,
              const float* __restrict__ bias,
              float* __restrict__ C, int ldc,
              int M, int N, int K) {
    __shared__ float Bs[(KC_ / 4) * PKSTRIDE_];  // 2560 floats = 10 KB
    const int tid  = threadIdx.x;
    const int wave = tid >> 5;
    const int lane = tid & 31;
    const int half = lane >> 4;
    const int l16  = lane & 15;
    const int m0 = blockIdx.x * 128 + wave * 32;  // two 16-row tiles: m0, m0+16
    const int n0 = blockIdx.y * 64;

    const float* Arow0 = A + (size_t)(m0 + l16) * lda;
    const float* Arow1 = A + (size_t)(m0 + 16 + l16) * lda;
    const float* BsBase = &Bs[half * 160 + l16 * 2];  // + p*320 + t*32 (imm)

    v8f acc[2][4];
#pragma unroll
    for (int mi = 0; mi < 2; ++mi)
#pragma unroll
        for (int t = 0; t < 4; ++t) acc[mi][t] = (v8f){};

    for (int kchunk = 0; kchunk < K; kchunk += KC_) {
        __syncthreads();
        if (BT) {
            // B element (k,n) = Bm[(n0+n)*ldb + kchunk + k]; re-pack to fragment order
            const int n  = tid >> 1;           // 0..63
            const int kq = (tid & 1) * 16;     // 0 or 16
            const float* src = Bm + (size_t)(n0 + n) * ldb + kchunk + kq;
#pragma unroll
            for (int i = 0; i < 4; ++i) {
                v4f v = *(const v4f*)(src + i * 4);
                const int p = (kq + i * 4) >> 2;
                float* dst = &Bs[p * PKSTRIDE_ + n * 2];
                *(v2f*)(dst)       = (v2f){v.x, v.z};  // s=0: {k, k+2}
                *(v2f*)(dst + 160) = (v2f){v.y, v.w};  // s=1: {k+1, k+3}
            }
        } else {
            const int kr = tid >> 2;           // K row 0..31
            const int c0 = (tid & 3) * 16;     // 0,16,32,48
            const int p  = kr >> 2;
            const int s  = kr & 1;
            const int sl = (kr & 3) >> 1;      // pair slot
            float* dst = &Bs[p * PKSTRIDE_ + s * 160 + sl];
            const float* src = Bm + (size_t)(kchunk + kr) * ldb;
            if (n0 + 64 <= N) {
#pragma unroll
                for (int i = 0; i < 4; ++i) {
                    v4f v = *(const v4f*)(src + n0 + c0 + i * 4);
                    const int nb = c0 + i * 4;
                    dst[(nb + 0) * 2] = v.x;
                    dst[(nb + 1) * 2] = v.y;
                    dst[(nb + 2) * 2] = v.z;
                    dst[(nb + 3) * 2] = v.w;
                }
            } else {
#pragma unroll
                for (int j = 0; j < 16; ++j) {
                    int col = n0 + c0 + j;
                    int cc = col < N ? col : N - 1;  // clamp, no OOB / EXEC games
                    dst[(c0 + j) * 2] = src[cc];
                }
            }
        }
        __syncthreads();

        // speculative prefetch of next B chunk (global_prefetch_b8)
        if (kchunk + KC_ < K) {
            const float* pf = BT
                ? Bm + (size_t)(n0 + (tid >> 1)) * ldb + kchunk + KC_
                : Bm + (size_t)(kchunk + KC_ + (tid >> 2)) * ldb + n0;
            __builtin_prefetch(pf, 0, 3);
        }

#pragma unroll
        for (int kc = 0; kc < KC_; kc += 4) {
            const int p = kc >> 2;
            const int ka = kchunk + kc + 2 * half;
            v2f a0 = *(const v2f*)(Arow0 + ka);
            v2f a1 = *(const v2f*)(Arow1 + ka);
#pragma unroll
            for (int t = 0; t < 4; ++t) {
                // one ds_load_b64, already in WMMA B-fragment order
                v2f b = *(const v2f*)(BsBase + p * PKSTRIDE_ + t * 32);
                acc[0][t] = __builtin_amdgcn_wmma_f32_16x16x4_f32(
                    false, a0, false, b, (short)0, acc[0][t], false, false);
                acc[1][t] = __builtin_amdgcn_wmma_f32_16x16x4_f32(
                    false, a1, false, b, (short)0, acc[1][t], false, false);
            }
        }
    }

    // epilogue: D layout — VGPR r: lanes 0-15 -> (m0+r, l16), lanes 16-31 -> (m0+r+8)
#pragma unroll
    for (int mi = 0; mi < 2; ++mi) {
#pragma unroll
        for (int t = 0; t < 4; ++t) {
            int col = n0 + t * 16 + l16;
            if (col < N) {
                float bv = (MODE == 2) ? 0.0f : bias[col];
#pragma unroll
                for (int r = 0; r < 8; ++r) {
                    int row = m0 + mi * 16 + r + 8 * half;
                    float v = acc[mi][t][r] + bv;
                    if (MODE == 1) v = tanhf(v);
                    C[(size_t)row * ldc + col] = v;
                }
            }
        }
    }
}

// ---------------- row L2 normalize ----------------
__global__ void l2norm_rows_kernel(const float* __restrict__ in,
                                   float* __restrict__ out, int cols) {
    const int wave = threadIdx.x >> 5;
    const int lane = threadIdx.x & 31;
    const int r = blockIdx.x * 8 + wave;
    const float* src = in + (size_t)r * cols;
    float* dst = out + (size_t)r * cols;
    float ss = 0.0f;
    for (int c = lane; c < cols; c += 32) {
        float v = src[c];
        ss += v * v;
    }
    ss = waveSum(ss);
    float inv = 1.0f / fmaxf(sqrtf(ss), EPS_);
    for (int c = lane; c < cols; c += 32) dst[c] = src[c] * inv;
}

// ---------------- row inverse-norm (strided rows) ----------------
__global__ void rownorm_inv_kernel(const float* __restrict__ in, int stride,
                                   int cols, float* __restrict__ invn) {
    const int wave = threadIdx.x >> 5;
    const int lane = threadIdx.x & 31;
    const int r = blockIdx.x * 8 + wave;
    const float* src = in + (size_t)r * stride;
    float ss = 0.0f;
    for (int c = lane; c < cols; c += 32) {
        float v = src[c];
        ss += v * v;
    }
    ss = waveSum(ss);
    if (lane == 0) invn[r] = 1.0f / fmaxf(sqrtf(ss), EPS_);
}

// ---------------- CE over logits (512 x 40) ----------------
__global__ void ce_logits_kernel(const float* __restrict__ logits,
                                 const int* __restrict__ labels,
                                 float* __restrict__ out) {
    const int i = blockIdx.x;
    const int l = threadIdx.x;  // 32 threads
    const float* row = logits + (size_t)i * NL_;
    float x0 = (l < NL_) ? row[l] : -3.0e38f;
    float x1 = (l + 32 < NL_) ? row[l + 32] : -3.0e38f;
    float m = waveMax(fmaxf(x0, x1));
    float e = ((l < NL_) ? __expf(x0 - m) : 0.0f) +
              ((l + 32 < NL_) ? __expf(x1 - m) : 0.0f);
    float s = waveSum(e);
    if (l == 0) {
        float loss = m + logf(s) - row[labels[i]];
        atomicAdd(out, loss * (1.0f / B_));
    }
}

// ---------------- margin loss over C (512 x 512) ----------------
__global__ void margin_kernel(const float* __restrict__ Cm,
                              const int* __restrict__ labels,
                              float* __restrict__ out) {
    __shared__ float sm[8];
    const int i = blockIdx.x;
    const int li = labels[i];
    float m = -3.0e38f;
    for (int j = threadIdx.x; j < B_; j += 256)
        if (labels[j] != li) m = fmaxf(m, Cm[(size_t)i * B_ + j]);
    m = blockMax(m, sm);
    if (threadIdx.x == 0) {
        float pos = Cm[(size_t)i * B_ + i];
        float neg = fmaxf(m, 0.0f);
        float v = fmaxf(neg - pos + GAMMA_, 0.0f);
        atomicAdd(out, v * (1.0f - ALPHA_));
    }
}

// ---------------- contrastive CE over cos (512 x 512, scaled on the fly) ----------------
__global__ void cl_loss_kernel(const float* __restrict__ dots,
                               const float* __restrict__ invn1,
                               const float* __restrict__ invn2,
                               float* __restrict__ out) {
    __shared__ float sm[8];
    const int i = blockIdx.x;
    const float a = invn1[i] * (1.0f / TEMP_);
    float m = -3.0e38f;
    for (int j = threadIdx.x; j < B_; j += 256)
        m = fmaxf(m, dots[(size_t)i * B_ + j] * a * invn2[j]);
    m = blockMax(m, sm);
    __syncthreads();
    float s = 0.0f;
    for (int j = threadIdx.x; j < B_; j += 256)
        s += __expf(dots[(size_t)i * B_ + j] * a * invn2[j] - m);
    s = blockSum(s, sm);
    if (threadIdx.x == 0) {
        float sii = dots[(size_t)i * B_ + i] * a * invn2[i];
        float loss = m + logf(s) - sii;
        atomicAdd(out, ALPHA_ * loss * (1.0f / B_));
    }
}

// ---------------- workspace layout (floats) ----------------
#define OFF_RH     ((size_t)0)                         // 512*2304
#define OFF_TOKIN  (OFF_RH    + (size_t)B_ * H3_)      // 2048*768
#define OFF_TOKS   (OFF_TOKIN + (size_t)B_ * 4 * H_)   // 2048*768
#define OFF_LOGITS (OFF_TOKS  + (size_t)B_ * 4 * H_)   // 512*40
#define OFF_RELE   (OFF_LOGITS+ (size_t)B_ * NL_)      // 512*384
#define OFF_RELN   (OFF_RELE  + (size_t)B_ * DS_)      // 512*384
#define OFF_DESCN  (OFF_RELN  + (size_t)B_ * DS_)      // 512*384
#define OFF_CMAT   (OFF_DESCN + (size_t)B_ * DS_)      // 512*512
#define OFF_COS    (OFF_CMAT  + (size_t)B_ * B_)       // 512*512
#define OFF_N1     (OFF_COS   + (size_t)B_ * B_)       // 512
#define OFF_N2     (OFF_N1    + (size_t)B_)            // 512

extern "C" void kernel_launch(void* const* d_in, const int* in_sizes, int n_in,
                              void* d_out, int out_size, void* d_ws, size_t ws_size,
                              hipStream_t stream) {
    const float* bert    = (const float*)d_in[0];
    const float* desc    = (const float*)d_in[1];
    const float* dense_w = (const float*)d_in[2];
    const float* dense_b = (const float*)d_in[3];
    const float* cls_w   = (const float*)d_in[4];
    const float* cls_b   = (const float*)d_in[5];
    const float* fc_w    = (const float*)d_in[6];
    const float* fc_b    = (const float*)d_in[7];
    const int*   eidx    = (const int*)d_in[8];
    const int*   labels  = (const int*)d_in[9];
    float* out = (float*)d_out;
    float* ws  = (float*)d_ws;

    float* RH     = ws + OFF_RH;
    float* TOKIN  = ws + OFF_TOKIN;
    float* TOKS   = ws + OFF_TOKS;
    float* LOGITS = ws + OFF_LOGITS;
    float* RELE   = ws + OFF_RELE;
    float* RELN   = ws + OFF_RELN;
    float* DESCN  = ws + OFF_DESCN;
    float* CMAT   = ws + OFF_CMAT;
    float* COS    = ws + OFF_COS;
    float* N1     = ws + OFF_N1;
    float* N2     = ws + OFF_N2;

    init_out_kernel<<<1, 32, 0, stream>>>(out);

    // gather + tanh prep
    gather_prep_kernel<<<B_, 256, 0, stream>>>(bert, eidx, RH, TOKIN);

    // logits = RH(512x2304) @ cls_w(2304x40) + cls_b
    gemm_wmma_f32<0, false><<<dim3(B_ / 128, 1), 128, 0, stream>>>(
        RH, H3_, cls_w, NL_, cls_b, LOGITS, NL_, B_, NL_, H3_);

    // rel = RH @ fc_w(2304x384) + fc_b
    gemm_wmma_f32<0, false><<<dim3(B_ / 128, DS_ / 64), 128, 0, stream>>>(
        RH, H3_, fc_w, DS_, fc_b, RELE, DS_, B_, DS_, H3_);

    // toks = tanh(TOKIN(2048x768) @ dense_w(768x768) + dense_b)
    gemm_wmma_f32<1, false><<<dim3((B_ * 4) / 128, H_ / 64), 128, 0, stream>>>(
        TOKIN, H_, dense_w, H_, dense_b, TOKS, H_, B_ * 4, H_, H_);

    // row-normalize rel and desc
    l2norm_rows_kernel<<<B_ / 8, 256, 0, stream>>>(RELE, RELN, DS_);
    l2norm_rows_kernel<<<B_ / 8, 256, 0, stream>>>(desc, DESCN, DS_);

    // C = RELN(512x384) @ DESCN^T  (NT)
    gemm_wmma_f32<2, true><<<dim3(B_ / 128, B_ / 64), 128, 0, stream>>>(
        RELN, DS_, DESCN, DS_, nullptr, CMAT, B_, B_, B_, DS_);

    // z1/z2 inverse norms (contiguous 1536-length strided views of TOKS)
    rownorm_inv_kernel<<<B_ / 8, 256, 0, stream>>>(TOKS, 4 * H_, 2 * H_, N1);
    rownorm_inv_kernel<<<B_ / 8, 256, 0, stream>>>(TOKS + 2 * H_, 4 * H_, 2 * H_, N2);

    // raw dots = Z1(512x1536, ld 3072) @ Z2^T (NT)
    gemm_wmma_f32<2, true><<<dim3(B_ / 128, B_ / 64), 128, 0, stream>>>(
        TOKS, 4 * H_, TOKS + 2 * H_, 4 * H_, nullptr, COS, B_, B_, B_, 2 * H_);

    // losses -> scalar
    ce_logits_kernel<<<B_, 32, 0, stream>>>(LOGITS, labels, out);
    margin_kernel<<<B_, 256, 0, stream>>>(CMAT, labels, out);
    cl_loss_kernel<<<B_, 256, 0, stream>>>(COS, N1, N2, out);
}